// SaccadicViTMultiStatePattern_62440234549861
// MI455X (gfx1250) — compile-verified
//
#include <hip/hip_runtime.h>
#include <math.h>

// ---- problem constants (match reference) ----
#define HD 768
#define CD 64
#define NPAT 64
#define KPAT 3
#define BEAMW 16
#define NNODES 256
#define NCTX 256
#define BSZ 2
#define LDW 68   // LDS row stride in floats: 68*4=272 bytes -> 16B aligned rows, bank offset 4

typedef float v2f __attribute__((ext_vector_type(2)));
typedef float v8f __attribute__((ext_vector_type(8)));
typedef int v4i_b __attribute__((vector_size(16)));  // matches builtin param pointee

// ---- CDNA5 async global<->LDS path (ASYNCcnt), with safe fallback ----
#if __has_builtin(__builtin_amdgcn_global_load_async_to_lds_b128)
#define ASYNC_LDS 1
#define LDS_AS __attribute__((address_space(3)))
#define GLB_AS __attribute__((address_space(1)))
#else
#define ASYNC_LDS 0
#endif

// async (or plain) 16B global -> LDS copy, one lane
__device__ __forceinline__ void g2l_b128(float* l, const float* g) {
#if ASYNC_LDS
  __builtin_amdgcn_global_load_async_to_lds_b128((GLB_AS v4i_b*)g, (LDS_AS v4i_b*)l, 0, 0);
#else
  *(float4*)l = *(const float4*)g;
#endif
}
// async (or plain) 16B LDS -> global copy, one lane
__device__ __forceinline__ void l2g_b128(float* g, float* l) {
#if ASYNC_LDS
  __builtin_amdgcn_global_store_async_from_lds_b128((GLB_AS v4i_b*)g, (LDS_AS v4i_b*)l, 0, 0);
#else
  *(float4*)g = *(const float4*)l;
#endif
}
__device__ __forceinline__ void async_wait0() {
#if ASYNC_LDS
#if __has_builtin(__builtin_amdgcn_s_wait_asynccnt)
  __builtin_amdgcn_s_wait_asynccnt(0);
#else
  asm volatile("s_wait_asynccnt 0x0" ::: "memory");
#endif
#endif
}

__device__ __forceinline__ v8f v8zero() {
  v8f z;
#pragma unroll
  for (int i = 0; i < 8; ++i) z[i] = 0.f;
  return z;
}

// One 16x16 output tile, full K=64 chunk contraction with V_WMMA_F32_16X16X4_F32.
// amBase: LDS row-major [M][K] rows of the A operand (row = tile_m*16 + lane&15)
// bnBase: LDS row-major [N][K] rows of the (transposed) B operand (row = tile_n*16 + lane&15)
__device__ __forceinline__ void gemm_tile(const float* amBase, const float* bnBase,
                                          int lane, v8f& acc) {
  const int koff = 2 * (lane >> 4);
#pragma unroll
  for (int k0 = 0; k0 < 64; k0 += 4) {
    v2f a = *(const v2f*)(amBase + k0 + koff);
    v2f b = *(const v2f*)(bnBase + k0 + koff);
    acc = __builtin_amdgcn_wmma_f32_16x16x4_f32(false, a, false, b, (short)0, acc, false, false);
  }
}

// Write a 16x16 f32 C/D fragment into LDS (row stride LDW):
// VGPR r holds M = r (lanes 0-15) / r+8 (lanes 16-31), N = lane&15.
__device__ __forceinline__ void store_tile(float* out, int tm, int tn, int lane, v8f c) {
  const int m0 = tm * 16 + 8 * (lane >> 4);
  const int nn = tn * 16 + (lane & 15);
#pragma unroll
  for (int rr = 0; rr < 8; ++rr) out[(m0 + rr) * LDW + nn] = c[rr];
}

__device__ __forceinline__ const float* search_row(const float* hid, const float* ctx,
                                                   int b, int node) {
  return (node < NNODES) ? hid + ((size_t)b * NNODES + node) * HD
                         : ctx + ((size_t)b * NCTX + (node - NNODES)) * HD;
}

// ------------------- kernels -------------------

// per-node squared norms over the concat(hidden, context) search set
__global__ __launch_bounds__(256) void k_norms(const float* __restrict__ hid,
                                               const float* __restrict__ ctx,
                                               float* __restrict__ norms) {
  const int wave = threadIdx.x >> 5, lane = threadIdx.x & 31;
  const int idx = blockIdx.x * 8 + wave;  // 0 .. BSZ*512-1
  const int b = idx >> 9, node = idx & 511;
  const float* row = search_row(hid, ctx, b, node);
  float acc = 0.f;
  for (int i = lane; i < HD; i += 32) { float v = row[i]; acc += v * v; }
#pragma unroll
  for (int m = 16; m > 0; m >>= 1) acc += __shfl_xor(acc, m, 32);
  if (lane == 0) norms[idx] = acc;
}

__global__ void k_init(int* __restrict__ pat, int* __restrict__ ni, float* __restrict__ jlp) {
  int t = blockIdx.x * blockDim.x + threadIdx.x;
  if (t < BSZ * NPAT) {
    pat[t] = t & 63;
    jlp[t] = 0.f;
    for (int k = 0; k < KPAT; ++k) ni[t * KPAT + k] = -1;
  }
}

// Per slot: G = A^T A (WMMA), Jacobi eigensolve of G -> V, lambda; derived quantities.
__global__ __launch_bounds__(128) void k_eigen(
    const float* __restrict__ covBase, const float* __restrict__ meanBase,
    int bcast, int Sb, const float* __restrict__ lsp,
    float* __restrict__ VT, float* __restrict__ INVQ, float* __restrict__ NPDC,
    float* __restrict__ MUA, float* __restrict__ DET, float* __restrict__ MUN) {
  __shared__ __align__(16) float AcT[64 * LDW];  // A chunk transposed: [col][k]
  __shared__ __align__(16) float G[64 * LDW];
  __shared__ __align__(16) float Vm[64 * LDW];
  __shared__ float muC[64];
  __shared__ float cA[32], sA[32];
  __shared__ int pA[32], qA[32];
  __shared__ float detAcc;

  const int s = blockIdx.x;
  const int r = s % Sb;
  const float* A = covBase + (size_t)(bcast ? r : s) * HD * CD;
  const float* mu = meanBase + (size_t)(bcast ? r : s) * HD;
  const int t = threadIdx.x, lane = t & 31, wave = t >> 5;

  v8f acc[4];
  for (int i = 0; i < 4; ++i) acc[i] = v8zero();
  float muAreg = 0.f, munreg = 0.f;

  for (int kc = 0; kc < HD; kc += 64) {
    for (int rep = 0; rep < 8; ++rep) {
      int lin = t + rep * 128;                 // 1024 float4 / 128 threads
      int kr = lin >> 4, c4 = (lin & 15) << 2;
      float4 v = *(const float4*)(A + (size_t)(kc + kr) * CD + c4);
      AcT[(c4 + 0) * LDW + kr] = v.x;
      AcT[(c4 + 1) * LDW + kr] = v.y;
      AcT[(c4 + 2) * LDW + kr] = v.z;
      AcT[(c4 + 3) * LDW + kr] = v.w;
    }
    if (t < 64) muC[t] = mu[kc + t];
    if (kc + 64 < HD) __builtin_prefetch(A + (size_t)(kc + 64) * CD, 0, 0);
    __syncthreads();
    for (int ti = 0; ti < 4; ++ti) {
      int tile = wave * 4 + ti;
      // Gram: (m,k)=A[k][m]=AcT[m][k]; (k,n)=A[k][n]=AcT[n][k]
      gemm_tile(AcT + ((tile >> 2) * 16 + (lane & 15)) * LDW,
                AcT + ((tile & 3) * 16 + (lane & 15)) * LDW, lane, acc[ti]);
    }
    if (t < 64) {
      float aa = 0.f;
      for (int kr = 0; kr < 64; ++kr) aa += muC[kr] * AcT[t * LDW + kr];
      muAreg += aa;                             // mu^T A column t
    } else if (t == 64) {
      float nn = 0.f;
      for (int kr = 0; kr < 64; ++kr) nn += muC[kr] * muC[kr];
      munreg += nn;                             // ||mu||^2
    }
    __syncthreads();
  }
  for (int ti = 0; ti < 4; ++ti) {
    int tile = wave * 4 + ti;
    store_tile(G, tile >> 2, tile & 3, lane, acc[ti]);
  }
  for (int rep = 0; rep < 32; ++rep) {
    int lin = t + rep * 128;
    int i = lin >> 6, j = lin & 63;
    Vm[i * LDW + j] = (i == j) ? 1.f : 0.f;
  }
  __syncthreads();

  // parallel cyclic Jacobi (round-robin tournament pairing), 8 sweeps
  for (int sweep = 0; sweep < 8; ++sweep) {
    for (int rnd = 0; rnd < 63; ++rnd) {
      if (t < 32) {
        int p, q;
        if (t == 0) { p = 63; q = rnd % 63; }
        else { p = (rnd + t) % 63; q = (rnd + 63 - t) % 63; }
        float app = G[p * LDW + p], aqq = G[q * LDW + q], apq = G[p * LDW + q];
        float c = 1.f, sn = 0.f;
        if (fabsf(apq) > 1e-12f) {
          float tau = (aqq - app) / (2.f * apq);
          float tt = (tau >= 0.f ? 1.f : -1.f) / (fabsf(tau) + sqrtf(1.f + tau * tau));
          c = rsqrtf(1.f + tt * tt);
          sn = tt * c;
        }
        pA[t] = p; qA[t] = q; cA[t] = c; sA[t] = sn;
      }
      __syncthreads();
      {  // row update J^T G
        int pr = t >> 2, j0 = (t & 3) << 4;
        int p = pA[pr], q = qA[pr];
        float c = cA[pr], sn = sA[pr];
        for (int j = j0; j < j0 + 16; ++j) {
          float gp = G[p * LDW + j], gq = G[q * LDW + j];
          G[p * LDW + j] = c * gp - sn * gq;
          G[q * LDW + j] = sn * gp + c * gq;
        }
      }
      __syncthreads();
      {  // column update (.)J  + accumulate V <- V J
        int pr = t >> 2, i0 = (t & 3) << 4;
        int p = pA[pr], q = qA[pr];
        float c = cA[pr], sn = sA[pr];
        for (int i = i0; i < i0 + 16; ++i) {
          float gp = G[i * LDW + p], gq = G[i * LDW + q];
          G[i * LDW + p] = c * gp - sn * gq;
          G[i * LDW + q] = sn * gp + c * gq;
          float vp = Vm[i * LDW + p], vq = Vm[i * LDW + q];
          Vm[i * LDW + p] = c * vp - sn * vq;
          Vm[i * LDW + q] = sn * vp + c * vq;
        }
      }
      __syncthreads();
    }
  }

  if (t == 0) detAcc = 0.f;
  __syncthreads();
  const float ls = *lsp;
  const float g = __expf(-ls);
  if (t < 64) {
    float lam = G[t * LDW + t];
    if (lam < 0.f) lam = 0.f;
    float lamh = lam * g;
    float inv = rsqrtf(1.f + lamh);
    INVQ[(size_t)s * CD + t] = inv;        // Hmat column scale
    NPDC[(size_t)s * CD + t] = g * inv;    // npd column scale
    MUA[(size_t)s * CD + t] = muAreg;
    atomicAdd(&detAcc, log1pf(lamh));
  }
  if (t == 64) MUN[s] = munreg;
  __syncthreads();
  if (t == 0) DET[s] = -0.5f * ((float)HD * ls + detAcc);
  for (int rep = 0; rep < 32; ++rep) {
    int lin = t + rep * 128;
    int cc = lin >> 6, jj = lin & 63;
    VT[(size_t)s * (CD * CD) + lin] = Vm[jj * LDW + cc];  // VT[c][j] = V[j][c]
  }
}

// Per (slot, 64-node block): y = S@A (WMMA, K=768), z = (y - muA)@V (WMMA, K=64), scores.
__global__ __launch_bounds__(256) void k_scores(
    const float* __restrict__ hid, const float* __restrict__ ctx,
    const float* __restrict__ covBase, const float* __restrict__ meanBase,
    int bcast, int Sb, int n, int N, const float* __restrict__ lsp,
    const float* __restrict__ VT, const float* __restrict__ NPDC,
    const float* __restrict__ MUA, const float* __restrict__ DET,
    const float* __restrict__ MUN, const float* __restrict__ norms,
    const float* __restrict__ jlpIn, float* __restrict__ scores) {
  __shared__ __align__(16) float bufA[64 * LDW];  // S chunk -> w -> z
  __shared__ __align__(16) float bufB[64 * LDW];  // A^T chunk -> VT
  __shared__ float muC[64], muAl[64], dmu[64], qld[64];

  const int s = blockIdx.x;
  const int b = s / Sb, r = s % Sb, beam = r / n;
  const int node0 = blockIdx.y * 64;
  const float* A = covBase + (size_t)(bcast ? r : s) * HD * CD;
  const float* mu = meanBase + (size_t)(bcast ? r : s) * HD;
  const int t = threadIdx.x, lane = t & 31, wave = t >> 5;
  const int tile0 = wave * 2, tile1 = tile0 + 1;
  const int tm0 = tile0 >> 2, tn0 = tile0 & 3, tm1 = tile1 >> 2, tn1 = tile1 & 3;

  if (t < 64) {
    dmu[t] = 0.f;
    muAl[t] = MUA[(size_t)s * CD + t];
    float nc = NPDC[(size_t)s * CD + t];
    qld[t] = nc * nc;
  }
  v8f acc0 = v8zero(), acc1 = v8zero();

  for (int kc = 0; kc < HD; kc += 64) {
    for (int rep = 0; rep < 4; ++rep) {     // S chunk row-major [node][k]: async DMA to LDS
      int lin = t + rep * 256;
      int row = lin >> 4, k4 = (lin & 15) << 2;
      const float* rp = search_row(hid, ctx, b, node0 + row);
      g2l_b128(bufA + row * LDW + k4, rp + kc + k4);
    }
    for (int rep = 0; rep < 4; ++rep) {     // A chunk transposed [col][k] (scatter: VALU path)
      int lin = t + rep * 256;
      int kr = lin >> 4, c4 = (lin & 15) << 2;
      float4 v = *(const float4*)(A + (size_t)(kc + kr) * CD + c4);
      bufB[(c4 + 0) * LDW + kr] = v.x;
      bufB[(c4 + 1) * LDW + kr] = v.y;
      bufB[(c4 + 2) * LDW + kr] = v.z;
      bufB[(c4 + 3) * LDW + kr] = v.w;
    }
    if (t < 64) muC[t] = mu[kc + t];
    if (kc + 64 < HD) __builtin_prefetch(A + (size_t)(kc + 64) * CD, 0, 0);
    async_wait0();
    __syncthreads();
    gemm_tile(bufA + (tm0 * 16 + (lane & 15)) * LDW, bufB + (tn0 * 16 + (lane & 15)) * LDW, lane, acc0);
    gemm_tile(bufA + (tm1 * 16 + (lane & 15)) * LDW, bufB + (tn1 * 16 + (lane & 15)) * LDW, lane, acc1);
    if (t < 64) {                           // s . mu partial
      float dd = 0.f;
      for (int kr = 0; kr < 64; ++kr) dd += bufA[t * LDW + kr] * muC[kr];
      dmu[t] += dd;
    }
    __syncthreads();
  }
  {  // w = y - muA -> bufA
    int kh = lane >> 4, nl = lane & 15;
#pragma unroll
    for (int rr = 0; rr < 8; ++rr) {
      int m0 = tm0 * 16 + 8 * kh + rr, n0c = tn0 * 16 + nl;
      bufA[m0 * LDW + n0c] = acc0[rr] - muAl[n0c];
      int m1 = tm1 * 16 + 8 * kh + rr, n1c = tn1 * 16 + nl;
      bufA[m1 * LDW + n1c] = acc1[rr] - muAl[n1c];
    }
  }
  for (int rep = 0; rep < 16; ++rep) {      // VT -> bufB
    int lin = t + rep * 256;
    bufB[(lin >> 6) * LDW + (lin & 63)] = VT[(size_t)s * (CD * CD) + lin];
  }
  __syncthreads();
  v8f z0 = v8zero(), z1 = v8zero();
  gemm_tile(bufA + (tm0 * 16 + (lane & 15)) * LDW, bufB + (tn0 * 16 + (lane & 15)) * LDW, lane, z0);
  gemm_tile(bufA + (tm1 * 16 + (lane & 15)) * LDW, bufB + (tn1 * 16 + (lane & 15)) * LDW, lane, z1);
  __syncthreads();
  store_tile(bufA, tm0, tn0, lane, z0);
  store_tile(bufA, tm1, tn1, lane, z1);
  __syncthreads();
  if (t < 64) {
    float accn = 0.f;
    for (int c = 0; c < 64; ++c) {
      float zz = bufA[t * LDW + c];
      accn += zz * zz * qld[c];
    }
    const float ls = *lsp;
    const float g = __expf(-ls);
    const int node = node0 + t;
    float sn = norms[b * 512 + node];
    float val = jlpIn[b * NPAT + beam] + (-0.5f * (float)HD * 1.8378770664093453f) + DET[s]
                - 0.5f * (g * (sn - 2.f * dmu[t] + MUN[s]) - accn);
    scores[((size_t)b * Sb + r) * N + node] = val;
  }
}

// top-16 per batch (iterative argmax + mask; lowest-index tie break like jax top_k)
__global__ __launch_bounds__(256) void k_topk(float* __restrict__ scores, int Sb, int n, int N,
                                              float* __restrict__ vals, int* __restrict__ selBeam,
                                              int* __restrict__ selSlot, int* __restrict__ selNode) {
  __shared__ float bv[256];
  __shared__ int bi[256];
  const int b = blockIdx.x, t = threadIdx.x;
  const int len = Sb * N;
  float* base = scores + (size_t)b * Sb * N;
  for (int kk = 0; kk < BEAMW; ++kk) {
    float best = -INFINITY;
    int bidx = 0x7fffffff;
    for (int i = t; i < len; i += 256) {
      float v = base[i];
      if (v > best || (v == best && i < bidx)) { best = v; bidx = i; }
    }
    bv[t] = best; bi[t] = bidx;
    __syncthreads();
    for (int st = 128; st > 0; st >>= 1) {
      if (t < st) {
        float v2 = bv[t + st]; int i2 = bi[t + st];
        if (v2 > bv[t] || (v2 == bv[t] && i2 < bi[t])) { bv[t] = v2; bi[t] = i2; }
      }
      __syncthreads();
    }
    if (t == 0) {
      int fi = bi[0];
      vals[b * BEAMW + kk] = bv[0];
      selBeam[b * BEAMW + kk] = fi / (n * N);
      selSlot[b * BEAMW + kk] = (fi / N) % n;
      selNode[b * BEAMW + kk] = fi % N;
      base[fi] = -INFINITY;
    }
    __threadfence();
    __syncthreads();
  }
}

// Beam bookkeeping + sel_npd + new_cov = cov_u @ Hmat (WMMA) + new_mean.
__global__ __launch_bounds__(256) void k_update(
    const float* __restrict__ hid, const float* __restrict__ ctx,
    const float* __restrict__ covBase, const float* __restrict__ meanBase,
    int bcast, int Sb, int n,
    const float* __restrict__ VT, const float* __restrict__ INVQ, const float* __restrict__ NPDC,
    const float* __restrict__ tvals, const int* __restrict__ tbeam,
    const int* __restrict__ tslot, const int* __restrict__ tnode,
    const int* __restrict__ patIn, const int* __restrict__ niIn,
    int* __restrict__ patOut, int* __restrict__ niOut, float* __restrict__ jlpOut,
    float* __restrict__ covOut, float* __restrict__ meanOut,
    float* __restrict__ dout, int final_it) {
  __shared__ __align__(16) float HmT[64 * LDW];   // Hmat transposed: [c][k]
  __shared__ __align__(16) float chunkS[64 * LDW];
  __shared__ __align__(16) float outc[64 * LDW];
  __shared__ float uL[64], npdL[64];
  const int b = blockIdx.x / BEAMW, j = blockIdx.x % BEAMW;
  const int t = threadIdx.x, lane = t & 31, wave = t >> 5;
  const int beam = tbeam[b * BEAMW + j], slot = tslot[b * BEAMW + j], node = tnode[b * BEAMW + j];
  const float val = tvals[b * BEAMW + j];
  const int rsel = beam * n + slot, ssel = b * Sb + rsel;

  if (t == 0) {
    int pv = patIn[b * NPAT + beam];
    patOut[b * NPAT + j] = pv;
    int arr[KPAT];
    for (int k = 0; k < KPAT; ++k) arr[k] = niIn[(b * NPAT + beam) * KPAT + k];
    int cnt = 0;
    for (int k = 0; k < KPAT; ++k)
      if (arr[k] == -1) { if (cnt == slot) arr[k] = node; ++cnt; }
    for (int k = 0; k < KPAT; ++k) niOut[(b * NPAT + j) * KPAT + k] = arr[k];
    jlpOut[b * NPAT + j] = val;
    if (final_it) {  // d_out: [pat(2,16) | ni(2,16,3) | vals(2,16)] as float
      dout[b * BEAMW + j] = (float)pv;
      for (int k = 0; k < KPAT; ++k)
        dout[BSZ * BEAMW + (b * BEAMW + j) * KPAT + k] = (float)arr[k];
      dout[BSZ * BEAMW + BSZ * BEAMW * KPAT + b * BEAMW + j] = val;
    }
  }
  if (n == 1) return;  // uniform: last iteration has no remaining slots

  // sel_npd = ((s - mu) @ A @ V) * (g / sqrt(1 + lam*g))
  const float* Asel = covBase + (size_t)(bcast ? rsel : ssel) * HD * CD;
  const float* musel = meanBase + (size_t)(bcast ? rsel : ssel) * HD;
  const float* snode = search_row(hid, ctx, b, node);
  if (t < 64) {
    float u = 0.f;
    for (int k = 0; k < HD; ++k) u += (snode[k] - musel[k]) * Asel[(size_t)k * CD + t];
    uL[t] = u;
  }
  __syncthreads();
  if (t < 64) {
    float z = 0.f;
    const float* vtrow = VT + (size_t)ssel * (CD * CD) + (size_t)t * CD;
    for (int jj = 0; jj < 64; ++jj) z += uL[jj] * vtrow[jj];
    npdL[t] = z * NPDC[(size_t)ssel * CD + t];
  }
  for (int rep = 0; rep < 16; ++rep) {      // HmT[c][k] = V[k][c] * invsq[c]
    int lin = t + rep * 256;
    int cc = lin >> 6, kk = lin & 63;
    HmT[cc * LDW + kk] = VT[(size_t)ssel * (CD * CD) + lin] * INVQ[(size_t)ssel * CD + cc];
  }
  __syncthreads();

  const int tile0 = wave * 2, tile1 = tile0 + 1;
  const int tm0 = tile0 >> 2, tn0 = tile0 & 3, tm1 = tile1 >> 2, tn1 = tile1 & 3;
  for (int ui = 0; ui < n - 1; ++ui) {
    const int us = ui + (ui >= slot ? 1 : 0);
    const int ru = beam * n + us, su = b * Sb + ru;
    const float* covU = covBase + (size_t)(bcast ? ru : su) * HD * CD;
    const float* meanU = meanBase + (size_t)(bcast ? ru : su) * HD;
    float* covO = covOut + ((size_t)(b * BEAMW + j) * (n - 1) + ui) * HD * CD;
    float* meanO = meanOut + ((size_t)(b * BEAMW + j) * (n - 1) + ui) * HD;
    for (int mc = 0; mc < HD; mc += 64) {
      for (int rep = 0; rep < 4; ++rep) {   // cov chunk: async DMA to LDS
        int lin = t + rep * 256;
        int row = lin >> 4, c4 = (lin & 15) << 2;
        g2l_b128(chunkS + row * LDW + c4, covU + (size_t)(mc + row) * CD + c4);
      }
      async_wait0();
      __syncthreads();
      v8f a0 = v8zero(), a1 = v8zero();
      gemm_tile(chunkS + (tm0 * 16 + (lane & 15)) * LDW, HmT + (tn0 * 16 + (lane & 15)) * LDW, lane, a0);
      gemm_tile(chunkS + (tm1 * 16 + (lane & 15)) * LDW, HmT + (tn1 * 16 + (lane & 15)) * LDW, lane, a1);
      store_tile(outc, tm0, tn0, lane, a0);
      store_tile(outc, tm1, tn1, lane, a1);
      __syncthreads();
      if (t < 64) {  // new_mean = mean_u + new_cov @ sel_npd
        float addv = 0.f;
        for (int c = 0; c < 64; ++c) addv += outc[t * LDW + c] * npdL[c];
        meanO[mc + t] = meanU[mc + t] + addv;
      }
      for (int rep = 0; rep < 4; ++rep) {   // new_cov rows: async DMA from LDS
        int lin = t + rep * 256;
        int row = lin >> 4, c4 = (lin & 15) << 2;
        l2g_b128(covO + (size_t)(mc + row) * CD + c4, outc + row * LDW + c4);
      }
      async_wait0();  // outc must be stable/consumed before next chunk overwrites it
      __syncthreads();
    }
  }
}

// ------------------- workspace layout -------------------
static constexpr size_t AL(size_t x) { return (x + 255) & ~size_t(255); }
static constexpr size_t SMAX = (size_t)BSZ * NPAT * KPAT;  // 384
static constexpr size_t O_NORMS = 0;
static constexpr size_t O_PAT0 = AL(O_NORMS + (size_t)BSZ * 512 * 4);
static constexpr size_t O_NI0  = AL(O_PAT0 + (size_t)BSZ * NPAT * 4);
static constexpr size_t O_JLP0 = AL(O_NI0 + (size_t)BSZ * NPAT * KPAT * 4);
static constexpr size_t O_PAT1 = AL(O_JLP0 + (size_t)BSZ * NPAT * 4);
static constexpr size_t O_NI1  = AL(O_PAT1 + (size_t)BSZ * NPAT * 4);
static constexpr size_t O_JLP1 = AL(O_NI1 + (size_t)BSZ * NPAT * KPAT * 4);
static constexpr size_t O_TV   = AL(O_JLP1 + (size_t)BSZ * NPAT * 4);
static constexpr size_t O_TB   = AL(O_TV + (size_t)BSZ * BEAMW * 4);
static constexpr size_t O_TS   = AL(O_TB + (size_t)BSZ * BEAMW * 4);
static constexpr size_t O_TN   = AL(O_TS + (size_t)BSZ * BEAMW * 4);
static constexpr size_t O_VT   = AL(O_TN + (size_t)BSZ * BEAMW * 4);
static constexpr size_t O_INV  = AL(O_VT + SMAX * CD * CD * 4);
static constexpr size_t O_NPDC = AL(O_INV + SMAX * CD * 4);
static constexpr size_t O_MUA  = AL(O_NPDC + SMAX * CD * 4);
static constexpr size_t O_DET  = AL(O_MUA + SMAX * CD * 4);
static constexpr size_t O_MUN  = AL(O_DET + SMAX * 4);
static constexpr size_t O_SC   = AL(O_MUN + SMAX * 4);
static constexpr size_t O_COVA = AL(O_SC + (size_t)BSZ * 192 * 512 * 4);
static constexpr size_t O_MEANA = AL(O_COVA + (size_t)BSZ * BEAMW * 2 * HD * CD * 4);
static constexpr size_t O_COVB = AL(O_MEANA + (size_t)BSZ * BEAMW * 2 * HD * 4);
static constexpr size_t O_MEANB = AL(O_COVB + (size_t)BSZ * BEAMW * 1 * HD * CD * 4);

extern "C" void kernel_launch(void* const* d_in, const int* in_sizes, int n_in,
                              void* d_out, int out_size, void* d_ws, size_t ws_size,
                              hipStream_t stream) {
  (void)in_sizes; (void)n_in; (void)out_size; (void)ws_size;
  const float* hid = (const float*)d_in[0];
  const float* ctx = (const float*)d_in[1];
  const float* mean3 = (const float*)d_in[2];
  const float* std3 = (const float*)d_in[3];
  const float* lsp = (const float*)d_in[4];
  float* out = (float*)d_out;
  char* ws = (char*)d_ws;

  float* norms = (float*)(ws + O_NORMS);
  int* pat0 = (int*)(ws + O_PAT0);  int* ni0 = (int*)(ws + O_NI0);  float* jlp0 = (float*)(ws + O_JLP0);
  int* pat1 = (int*)(ws + O_PAT1);  int* ni1 = (int*)(ws + O_NI1);  float* jlp1 = (float*)(ws + O_JLP1);
  float* tv = (float*)(ws + O_TV);
  int* tb = (int*)(ws + O_TB);  int* tsl = (int*)(ws + O_TS);  int* tnd = (int*)(ws + O_TN);
  float* VT = (float*)(ws + O_VT);
  float* INVQ = (float*)(ws + O_INV);
  float* NPDC = (float*)(ws + O_NPDC);
  float* MUA = (float*)(ws + O_MUA);
  float* DET = (float*)(ws + O_DET);
  float* MUN = (float*)(ws + O_MUN);
  float* scores = (float*)(ws + O_SC);
  float* covA = (float*)(ws + O_COVA);  float* meanA = (float*)(ws + O_MEANA);
  float* covB = (float*)(ws + O_COVB);  float* meanB = (float*)(ws + O_MEANB);

  k_norms<<<128, 256, 0, stream>>>(hid, ctx, norms);
  k_init<<<1, 128, 0, stream>>>(pat0, ni0, jlp0);

  // iteration 0: 64 beams, n=3 slots, N=256 (hidden only), cov/mean broadcast over batch
  k_eigen<<<384, 128, 0, stream>>>(std3, mean3, 1, 192, lsp, VT, INVQ, NPDC, MUA, DET, MUN);
  k_scores<<<dim3(384, 4), 256, 0, stream>>>(hid, ctx, std3, mean3, 1, 192, 3, 256, lsp,
                                             VT, NPDC, MUA, DET, MUN, norms, jlp0, scores);
  k_topk<<<BSZ, 256, 0, stream>>>(scores, 192, 3, 256, tv, tb, tsl, tnd);
  k_update<<<BSZ * BEAMW, 256, 0, stream>>>(hid, ctx, std3, mean3, 1, 192, 3,
                                            VT, INVQ, NPDC, tv, tb, tsl, tnd,
                                            pat0, ni0, pat1, ni1, jlp1, covA, meanA, out, 0);
  // iteration 1: 16 beams, n=2, N=512
  k_eigen<<<64, 128, 0, stream>>>(covA, meanA, 0, 32, lsp, VT, INVQ, NPDC, MUA, DET, MUN);
  k_scores<<<dim3(64, 8), 256, 0, stream>>>(hid, ctx, covA, meanA, 0, 32, 2, 512, lsp,
                                            VT, NPDC, MUA, DET, MUN, norms, jlp1, scores);
  k_topk<<<BSZ, 256, 0, stream>>>(scores, 32, 2, 512, tv, tb, tsl, tnd);
  k_update<<<BSZ * BEAMW, 256, 0, stream>>>(hid, ctx, covA, meanA, 0, 32, 2,
                                            VT, INVQ, NPDC, tv, tb, tsl, tnd,
                                            pat1, ni1, pat0, ni0, jlp0, covB, meanB, out, 0);
  // iteration 2: 16 beams, n=1, N=512 -> final outputs
  k_eigen<<<32, 128, 0, stream>>>(covB, meanB, 0, 16, lsp, VT, INVQ, NPDC, MUA, DET, MUN);
  k_scores<<<dim3(32, 8), 256, 0, stream>>>(hid, ctx, covB, meanB, 0, 16, 1, 512, lsp,
                                            VT, NPDC, MUA, DET, MUN, norms, jlp0, scores);
  k_topk<<<BSZ, 256, 0, stream>>>(scores, 16, 1, 512, tv, tb, tsl, tnd);
  k_update<<<BSZ * BEAMW, 256, 0, stream>>>(hid, ctx, covB, meanB, 0, 16, 1,
                                            VT, INVQ, NPDC, tv, tb, tsl, tnd,
                                            pat0, ni0, pat1, ni1, jlp1, covB, meanB, out, 1);
}